// EuclideanCodebook_51427938402904
// MI455X (gfx1250) — compile-verified
//
#include <hip/hip_runtime.h>
#include <math.h>

typedef __attribute__((ext_vector_type(2))) float v2f;
typedef __attribute__((ext_vector_type(8))) float v8f;

#define NPTS    32768
#define DIM     128
#define KCODES  2048
#define ROWS    16
#define KSTEPS  (DIM / 4)      // 32 wmma k-steps of 4
#define NTILES  (KCODES / 16)  // 128 column tiles
#define TEMP_INV (1.0f / 0.9f)
#define DIV_EPS  1e-6f

// ---------------------------------------------------------------------------
// ws layout (floats): [0, 2048)  = e_sq[k]
//                     [2048,4096)= class_sum[k] (global softmax-mass accumulator)
// ---------------------------------------------------------------------------
__global__ __launch_bounds__(256) void prep_kernel(const float* __restrict__ embed,
                                                   float* __restrict__ ws) {
    int k = blockIdx.x * blockDim.x + threadIdx.x;
    if (k < KCODES) {
        const float* e = embed + (size_t)k * DIM;
        float s = 0.f;
#pragma unroll 8
        for (int i = 0; i < DIM; ++i) s += e[i] * e[i];
        ws[k] = s;             // ||e_k||^2
        ws[KCODES + k] = 0.f;  // zero class accumulator
    }
}

// ---------------------------------------------------------------------------
// Main kernel: one workgroup = 16 points. 8 waves; each wave owns 16 column
// tiles and processes 4 of them concurrently (4 accumulator chains) so the
// scheduler can hide B-load latency and WMMA accumulation latency.
// A (the wave's 16 x-rows, 64 VGPRs) is loaded once and reused by all tiles.
// The 4 concurrent tiles are 128 codes apart = 64 KB in embed -> single base
// address + 24-bit immediate offsets for all 4 B streams.
// ---------------------------------------------------------------------------
__global__ __launch_bounds__(256) void codebook_main(
    const float* __restrict__ x, const float* __restrict__ embed,
    const float* __restrict__ e_sq, float* __restrict__ class_sum,
    float* __restrict__ q_out, float* __restrict__ ind_out)
{
    extern __shared__ float smem[];
    float* sc  = smem;                    // [ROWS][KCODES] score stripe
    float* cls = smem + ROWS * KCODES;    // [KCODES] per-WG class partial
    __shared__ int rowArg[ROWS];

    const int tid  = threadIdx.x;
    const int wave = tid >> 5;
    const int lane = tid & 31;
    const int half = lane >> 4;           // 0: dims k+0..1   1: dims k+2..3
    const int l16  = lane & 15;
    const int row0 = blockIdx.x * ROWS;

    for (int i = tid; i < KCODES; i += 256) cls[i] = 0.f;

    // ---- Phase 1: WMMA score tiles: s = 2*(x.e) - ||e||^2 -----------------
    // A (16x4 f32): lane l<16 holds row M=l dims kk+0..1; lanes 16-31 dims kk+2..3
    const float* aBase = x + (size_t)(row0 + l16) * DIM + half * 2;
    v2f aReg[KSTEPS];
#pragma unroll
    for (int kk = 0; kk < KSTEPS; ++kk)
        aReg[kk] = *(const v2f*)(aBase + kk * 4);

    // 4 iterations: wave handles tiles {t, t+8, t+16, t+24} concurrently
    for (int t = wave; t < NTILES; t += 32) {
        const int col0 = t * 16 + l16;    // chain j handles col0 + j*128
        const float* bBase = embed + (size_t)col0 * DIM + half * 2;

        v8f acc[4];
#pragma unroll
        for (int j = 0; j < 4; ++j)
            acc[j] = (v8f){0.f, 0.f, 0.f, 0.f, 0.f, 0.f, 0.f, 0.f};

#pragma unroll
        for (int kk = 0; kk < KSTEPS; ++kk) {
#pragma unroll
            for (int j = 0; j < 4; ++j) {
                // tile j is 8 tiles (=128 codes) further: +128*DIM floats
                v2f b = *(const v2f*)(bBase + kk * 4 + j * (128 * DIM));
                acc[j] = __builtin_amdgcn_wmma_f32_16x16x4_f32(
                    false, aReg[kk], false, b, (short)0, acc[j], false, false);
            }
        }

#pragma unroll
        for (int j = 0; j < 4; ++j) {
            const int col = col0 + j * 128;
            const float es = e_sq[col];
#pragma unroll
            for (int v = 0; v < 8; ++v) {
                // D layout: VGPR v holds row M = v (lanes 0-15) / v+8 (lanes 16-31)
                int r = v + half * 8;
                sc[r * KCODES + col] = 2.0f * acc[j][v] - es;
            }
        }
    }
    __syncthreads();

    // ---- Phase 2: per-row argmax + stable softmax (2 rows per wave) -------
    for (int rr = 0; rr < 2; ++rr) {
        const int r = wave * 2 + rr;
        float* srow = sc + r * KCODES;

        float mx = -3.4e38f; int arg = 0;
        for (int i = lane; i < KCODES; i += 32) {
            float v = srow[i];
            if (v > mx) { mx = v; arg = i; }
        }
#pragma unroll
        for (int off = 16; off > 0; off >>= 1) {
            float om = __shfl_xor(mx, off);
            int   oa = __shfl_xor(arg, off);
            if (om > mx || (om == mx && oa < arg)) { mx = om; arg = oa; }
        }

        float se = 0.f;
        for (int i = lane; i < KCODES; i += 32) {
            float e = __expf((srow[i] - mx) * TEMP_INV);
            srow[i] = e;                  // stash exp for prob pass
            se += e;
        }
#pragma unroll
        for (int off = 16; off > 0; off >>= 1) se += __shfl_xor(se, off);
        const float inv = 1.0f / se;

        for (int i = lane; i < KCODES; i += 32)
            atomicAdd(&cls[i], srow[i] * inv);   // ds_add_f32 (rows collide on i)

        if (lane == 0) {
            rowArg[r] = arg;
            ind_out[row0 + r] = (float)arg;
        }
    }
    __syncthreads();

    // ---- Phase 3: gather quantize rows ------------------------------------
    for (int idx = tid; idx < ROWS * DIM; idx += 256) {
        int r = idx >> 7, c = idx & (DIM - 1);
        q_out[(size_t)(row0 + r) * DIM + c] = embed[(size_t)rowArg[r] * DIM + c];
    }

    // ---- Phase 4: flush class partials to global accumulator --------------
    for (int i = tid; i < KCODES; i += 256)
        atomicAdd(&class_sum[i], cls[i]);        // global_atomic_add_f32
}

// ---------------------------------------------------------------------------
// Final reduction: diversity loss = sum_k p_k * log(p_k + eps), p_k = sum/N
// ---------------------------------------------------------------------------
__global__ __launch_bounds__(256) void loss_kernel(const float* __restrict__ class_sum,
                                                   float* __restrict__ loss_out) {
    __shared__ float red[256];
    float acc = 0.f;
    for (int i = threadIdx.x; i < KCODES; i += 256) {
        float cp = class_sum[i] * (1.0f / (float)NPTS);
        acc += cp * __logf(cp + DIV_EPS);
    }
    red[threadIdx.x] = acc;
    __syncthreads();
    for (int s = 128; s > 0; s >>= 1) {
        if (threadIdx.x < s) red[threadIdx.x] += red[threadIdx.x + s];
        __syncthreads();
    }
    if (threadIdx.x == 0) loss_out[0] = red[0];
}

// ---------------------------------------------------------------------------
extern "C" void kernel_launch(void* const* d_in, const int* in_sizes, int n_in,
                              void* d_out, int out_size, void* d_ws, size_t ws_size,
                              hipStream_t stream) {
    const float* x     = (const float*)d_in[0];  // [8,4096,128]
    const float* embed = (const float*)d_in[1];  // [2048,128]
    float* ws   = (float*)d_ws;
    float* out  = (float*)d_out;
    float* q_out    = out;                               // [32768*128]
    float* ind_out  = out + (size_t)NPTS * DIM;          // [32768]
    float* loss_out = ind_out + NPTS;                    // [1]

    prep_kernel<<<(KCODES + 255) / 256, 256, 0, stream>>>(embed, ws);

    size_t shmem = (size_t)(ROWS * KCODES + KCODES) * sizeof(float); // 136 KB < 320 KB/WGP
    codebook_main<<<NPTS / ROWS, 256, shmem, stream>>>(
        x, embed, ws /*e_sq*/, ws + KCODES /*class_sum*/, q_out, ind_out);

    loss_kernel<<<1, 256, 0, stream>>>(ws + KCODES, loss_out);
}